// AuraSTGNN_12884901888283
// MI455X (gfx1250) — compile-verified
//
#include <hip/hip_runtime.h>
#include <math.h>

typedef __attribute__((ext_vector_type(2))) float v2f;
typedef __attribute__((ext_vector_type(8))) float v8f;

// D = A(16x4 f32) * B(4x16 f32) + C(16x16 f32)
// 8-arg pattern: (neg_a, A, neg_b, B, c_mod, C, reuse_a, reuse_b)
#define WMMA_F32(a, b, c) \
  __builtin_amdgcn_wmma_f32_16x16x4_f32(false, (a), false, (b), (short)0, (c), false, false)

// ---------------------------------------------------------------- zero fill
__global__ void zero_f32(float* __restrict__ p, long n) {
  long i = (long)blockIdx.x * blockDim.x + threadIdx.x;
  long stride = (long)gridDim.x * blockDim.x;
  for (; i < n; i += stride) p[i] = 0.0f;
}

// ------------------------------------------------- edge scatter (mean agg)
// One thread = one float4 chunk of one edge's feature row.
template <int F, bool DO_DEG>
__global__ void scatter_add_k(const float* __restrict__ x,
                              const int* __restrict__ ei,
                              float* __restrict__ agg,
                              float* __restrict__ deg, int E) {
  constexpr int CH = F / 4;
  long t = (long)blockIdx.x * blockDim.x + threadIdx.x;
  long total = (long)E * CH;
  if (t >= total) return;
  int e = (int)(t / CH);
  int c = (int)(t % CH);
  int src = ei[e];
  int dst = ei[(long)E + e];
  const float4 v = ((const float4*)(x + (long)src * F))[c];
  float* a = agg + (long)dst * F + c * 4;
  unsafeAtomicAdd(a + 0, v.x);
  unsafeAtomicAdd(a + 1, v.y);
  unsafeAtomicAdd(a + 2, v.z);
  unsafeAtomicAdd(a + 3, v.w);
  if (DO_DEG && c == 0) unsafeAtomicAdd(deg + dst, 1.0f);
}

// ---------------------- fused SAGE layer: dual-GEMM + bias + ReLU + LayerNorm
// Wave handles 16 nodes. A tiles preloaded to VGPRs (ISA 16x4 f32 A layout:
// lane m (0-15) rows, lane>=16 carries K+2/K+3). D tile layout: vgpr j =
// row (j + 8*(lane>=16)), col = lane&15.
template <int K, int NOUT>
__global__ __launch_bounds__(256) void sage_gemm_ln(
    const float* __restrict__ xin, const float* __restrict__ agg,
    const float* __restrict__ deg, const float* __restrict__ Ws,
    const float* __restrict__ Wn, const float* __restrict__ bias,
    const float* __restrict__ gamma, const float* __restrict__ beta,
    float* __restrict__ out, int n_nodes) {
  __shared__ float lds[8][16][NOUT];
  const int lane = threadIdx.x & 31;
  const int wave = threadIdx.x >> 5;
  const int m = lane & 15;
  const int koff = (lane >> 4) * 2;  // 0 or 2
  const int nodeBase = blockIdx.x * 128 + wave * 16;
  int node = nodeBase + m;
  int nc = node < n_nodes ? node : n_nodes - 1;  // clamp (EXEC must stay full)
  float idg = 1.0f / fmaxf(deg[nc], 1.0f);

  const float* xr = xin + (long)nc * K;
  const float* ar = agg + (long)nc * K;
  v2f a_self[K / 4], a_neigh[K / 4];
#pragma unroll
  for (int kk = 0; kk < K / 4; ++kk) {
    int k = kk * 4 + koff;
    a_self[kk] = v2f{xr[k], xr[k + 1]};
    a_neigh[kk] = v2f{ar[k] * idg, ar[k + 1] * idg};
  }

  const int rbase = (lane >> 4) * 8;
#pragma unroll
  for (int nt = 0; nt < NOUT / 16; ++nt) {
    int col = nt * 16 + m;
    float bv = bias[col];
    v8f acc;
#pragma unroll
    for (int j = 0; j < 8; ++j) acc[j] = bv;
    const float* wsr = Ws + (long)col * K;
    const float* wnr = Wn + (long)col * K;
#pragma unroll
    for (int kk = 0; kk < K / 4; ++kk) {
      int k = kk * 4 + koff;
      v2f bs = {wsr[k], wsr[k + 1]};
      v2f bn = {wnr[k], wnr[k + 1]};
      acc = WMMA_F32(a_self[kk], bs, acc);
      acc = WMMA_F32(a_neigh[kk], bn, acc);
    }
#pragma unroll
    for (int j = 0; j < 8; ++j)
      lds[wave][rbase + j][col] = fmaxf(acc[j], 0.0f);
  }
  __syncthreads();

  // LayerNorm: 32 lanes cooperate per row (stride-32 columns, conflict-free)
  constexpr int ELEM = NOUT / 32;
  for (int r = 0; r < 16; ++r) {
    float v[ELEM];
    float s = 0.0f, sq = 0.0f;
#pragma unroll
    for (int i = 0; i < ELEM; ++i) {
      v[i] = lds[wave][r][i * 32 + lane];
      s += v[i];
      sq += v[i] * v[i];
    }
#pragma unroll
    for (int off = 16; off >= 1; off >>= 1) {
      s += __shfl_xor(s, off);
      sq += __shfl_xor(sq, off);
    }
    float mu = s * (1.0f / NOUT);
    float var = sq * (1.0f / NOUT) - mu * mu;
    float rstd = rsqrtf(var + 1e-5f);
    int rnode = nodeBase + r;
    if (rnode < n_nodes) {
#pragma unroll
      for (int i = 0; i < ELEM; ++i) {
        int c = i * 32 + lane;
        out[(long)rnode * NOUT + c] = (v[i] - mu) * rstd * gamma[c] + beta[c];
      }
    }
  }
}

// ----------------------------- MLP head: h[16x64] @ Wh1^T -> ReLU -> Wh2 -> sigmoid
__global__ __launch_bounds__(256) void head_kernel(
    const float* __restrict__ h, const float* __restrict__ Wh1,
    const float* __restrict__ bh1, const float* __restrict__ Wh2,
    const float* __restrict__ bh2, float* __restrict__ scores, int n_nodes) {
  const int lane = threadIdx.x & 31;
  const int wave = threadIdx.x >> 5;
  const int m = lane & 15;
  const int koff = (lane >> 4) * 2;
  const int nodeBase = blockIdx.x * 128 + wave * 16;
  int node = nodeBase + m;
  int nc = node < n_nodes ? node : n_nodes - 1;
  const float* hr = h + (long)nc * 64;
  const float* wr = Wh1 + (long)m * 64;

  float bv = bh1[m];
  v8f acc;
#pragma unroll
  for (int j = 0; j < 8; ++j) acc[j] = bv;
#pragma unroll
  for (int kk = 0; kk < 16; ++kk) {
    int k = kk * 4 + koff;
    v2f a = {hr[k], hr[k + 1]};
    v2f b = {wr[k], wr[k + 1]};
    acc = WMMA_F32(a, b, acc);
  }

  float w2 = Wh2[m];
  float bb = bh2[0];
  const int rbase = (lane >> 4) * 8;
#pragma unroll
  for (int j = 0; j < 8; ++j) {
    float t = fmaxf(acc[j], 0.0f) * w2;
    // reduce across the 16-lane half (xor masks < 16 stay in-half)
#pragma unroll
    for (int off = 8; off >= 1; off >>= 1) t += __shfl_xor(t, off);
    int rnode = nodeBase + rbase + j;
    if (m == 0 && rnode < n_nodes) {
      float z = t + bb;
      scores[rnode] = 1.0f / (1.0f + expf(-z));
    }
  }
}

// ---------------------------------------------------------------- launcher
extern "C" void kernel_launch(void* const* d_in, const int* in_sizes, int n_in,
                              void* d_out, int out_size, void* d_ws,
                              size_t ws_size, hipStream_t stream) {
  const float* x   = (const float*)d_in[0];
  const int*   ei  = (const int*)d_in[1];
  const float* Ws1 = (const float*)d_in[2];
  const float* Wn1 = (const float*)d_in[3];
  const float* b1  = (const float*)d_in[4];
  const float* g1  = (const float*)d_in[5];
  const float* be1 = (const float*)d_in[6];
  const float* Ws2 = (const float*)d_in[7];
  const float* Wn2 = (const float*)d_in[8];
  const float* b2  = (const float*)d_in[9];
  const float* g2  = (const float*)d_in[10];
  const float* be2 = (const float*)d_in[11];
  const float* Wh1 = (const float*)d_in[12];
  const float* bh1 = (const float*)d_in[13];
  const float* Wh2 = (const float*)d_in[14];
  const float* bh2 = (const float*)d_in[15];

  const int N = in_sizes[0] / 64;
  const int E = in_sizes[1] / 2;

  float* out    = (float*)d_out;
  float* scores = out;        // [N]
  float* h2     = out + N;    // [N,64]

  float* ws  = (float*)d_ws;
  float* deg = ws;                       // [N]
  float* h1  = deg + N;                  // [N,128]
  float* agg = h1 + (size_t)N * 128;     // [N,128] (layer1 uses first N*64 floats as [N,64])

  const int gemmBlocks = (N + 127) / 128;

  // Layer 1: mean-aggregate over 64 features, then fused GEMM+LN
  zero_f32<<<2048, 256, 0, stream>>>(deg, (long)N);
  zero_f32<<<2048, 256, 0, stream>>>(agg, (long)N * 64);
  {
    long total = (long)E * 16;
    scatter_add_k<64, true><<<(int)((total + 255) / 256), 256, 0, stream>>>(
        x, ei, agg, deg, E);
  }
  sage_gemm_ln<64, 128><<<gemmBlocks, 256, 0, stream>>>(
      x, agg, deg, Ws1, Wn1, b1, g1, be1, h1, N);

  // Layer 2: re-aggregate h1 (128 features), fused GEMM+LN into d_out's h slot
  zero_f32<<<2048, 256, 0, stream>>>(agg, (long)N * 128);
  {
    long total = (long)E * 32;
    scatter_add_k<128, false><<<(int)((total + 255) / 256), 256, 0, stream>>>(
        h1, ei, agg, nullptr, E);
  }
  sage_gemm_ln<128, 64><<<gemmBlocks, 256, 0, stream>>>(
      h1, agg, deg, Ws2, Wn2, b2, g2, be2, h2, N);

  // Head: scores = sigmoid(relu(h @ Wh1^T + bh1) @ Wh2^T + bh2)
  head_kernel<<<gemmBlocks, 256, 0, stream>>>(h2, Wh1, bh1, Wh2, bh2, scores, N);
}